// SparseConv3d_31628139167767
// MI455X (gfx1250) — compile-verified
//
#include <hip/hip_runtime.h>

typedef __attribute__((ext_vector_type(2))) float v2f;
typedef __attribute__((ext_vector_type(8))) float v8f;

#define B_    2
#define CIN   16
#define COUT  32
#define RES   64
#define VOX   (RES*RES*RES)
#define TAPS  27
#define WPB   8                          // waves per workgroup (wave32)
#define NROW  (B_*RES*RES)               // 8192 voxel rows (64 wide)
#define NBLK  (NROW/WPB)                 // 1024 workgroups

// LDS weight layout: v2f pairs of consecutive K-rows.
//   wl2[(tap*8 + q)*32 + n] = ( W[k = tap*16 + 2q][n], W[k = tap*16 + 2q+1][n] )
// B fragment for k-chunk s: rows (4s+2*hi, 4s+2*hi+1) -> q = 2s + hi, one b64 load.
#define WL2N  (TAPS*8*COUT)              // 6912 v2f = 54 KiB

__global__ __launch_bounds__(WPB*32) void sparse_conv3d_wmma_kernel(
    const float* __restrict__ x, const float* __restrict__ w,
    float* __restrict__ out)
{
    __shared__ v2f wl2[WL2N];
    const int tid = threadIdx.x;
    for (int d = tid; d < WL2N; d += WPB * 32) {
        const int n   = d & (COUT - 1);
        const int rp  = d >> 5;          // tap*8 + q
        const int tap = rp >> 3;
        const int q   = rp & 7;
        v2f t;
        t[0] = w[n * (CIN * TAPS) + (2 * q)     * TAPS + tap];
        t[1] = w[n * (CIN * TAPS) + (2 * q + 1) * TAPS + tap];
        wl2[d] = t;
    }
    __syncthreads();

    const int lane = tid & 31;
    const int wave = tid >> 5;
    const int hi   = lane >> 4;          // K-pair select within fragment
    const int lm   = lane & 15;          // voxel-in-segment (A) / out-channel (B,C)

    // One wave owns one full 64-voxel x-row: 4 segments x 32 out-channels.
    const int row = blockIdx.x * WPB + wave;     // 0..8191
    const int y = row & (RES - 1);
    const int z = (row >> 6) & (RES - 1);
    const int b = row >> 12;

    const float* xb = x + (size_t)b * CIN * VOX;

    // ---- Occupancy pre-pass (center voxels only, executed once) ----
    // Lane L and L+16 jointly cover all 16 input channels of voxel L%16.
    unsigned occ[4] = {0u, 0u, 0u, 0u};
    {
        const float* p0 = xb + (z * RES + y) * RES + lm;
        #pragma unroll
        for (int seg = 0; seg < 4; ++seg) {
            const float* p = p0 + seg * 16;
            #pragma unroll
            for (int s = 0; s < 4; ++s) {
                const int ci = s * 4 + 2 * hi;
                const float a0 = p[(size_t)ci * VOX];
                const float a1 = p[(size_t)(ci + 1) * VOX];
                occ[seg] |= ((a0 != 0.0f) | (a1 != 0.0f)) ? 1u : 0u;
            }
        }
    }

    v8f c[4][2] = {};                    // [segment][N-tile] accumulators (64 VGPRs)

    int tap = 0;
    for (int dz = 0; dz < 3; ++dz)
    for (int dy = 0; dy < 3; ++dy)
    for (int dx = 0; dx < 3; ++dx, ++tap) {
        const int zz = z + dz - 1;
        const int yy = y + dy - 1;
        const unsigned vzy = ((unsigned)zz < (unsigned)RES) &
                             ((unsigned)yy < (unsigned)RES);
        const int base_zy = (zz * RES + yy) * RES;

        bool valid[4];
        int  sp[4];
        #pragma unroll
        for (int seg = 0; seg < 4; ++seg) {
            const int xx = seg * 16 + lm + dx - 1;
            valid[seg] = (vzy & ((unsigned)xx < (unsigned)RES)) != 0u;
            sp[seg] = valid[seg] ? (base_zy + xx) : 0;
        }

        #pragma unroll
        for (int s = 0; s < 4; ++s) {
            // B fragments (shared by all 4 segments): one 2addr b64 LDS load
            const int bidx = (tap * 8 + s * 2 + hi) * COUT + lm;
            const v2f b0 = wl2[bidx];
            const v2f b1 = wl2[bidx + 16];

            const int ci = s * 4 + 2 * hi;   // A K-index pair base for this lane half

            #pragma unroll
            for (int seg = 0; seg < 4; ++seg) {
                const float* p = xb + sp[seg];
                float a0 = p[(size_t)ci * VOX];
                float a1 = p[(size_t)(ci + 1) * VOX];
                a0 = valid[seg] ? a0 : 0.0f;
                a1 = valid[seg] ? a1 : 0.0f;
                v2f a; a[0] = a0; a[1] = a1;

                c[seg][0] = __builtin_amdgcn_wmma_f32_16x16x4_f32(
                                false, a, false, b0, (short)0, c[seg][0], false, false);
                c[seg][1] = __builtin_amdgcn_wmma_f32_16x16x4_f32(
                                false, a, false, b1, (short)0, c[seg][1], false, false);
            }
        }
    }

    // ---- Apply occupancy mask and store ----
    #pragma unroll
    for (int seg = 0; seg < 4; ++seg) {
        const unsigned bal = __builtin_amdgcn_ballot_w32(occ[seg] != 0u);
        const unsigned m16 = (bal | (bal >> 16)) & 0xFFFFu;

        v8f d0 = c[seg][0], d1 = c[seg][1];
        #pragma unroll
        for (int r = 0; r < 8; ++r) {
            const float f = ((m16 >> (r + hi * 8)) & 1u) ? 1.0f : 0.0f;
            d0[r] *= f;
            d1[r] *= f;
        }

        // C/D layout: lane holds M = r + 8*hi -> 8 consecutive x positions
        const size_t sp_out = (size_t)z * RES * RES + (size_t)y * RES
                            + seg * 16 + hi * 8;
        float* dst0 = out + ((size_t)(b * COUT) + lm)      * VOX + sp_out;
        float* dst1 = out + ((size_t)(b * COUT) + lm + 16) * VOX + sp_out;
        *(float4*)(dst0)     = make_float4(d0[0], d0[1], d0[2], d0[3]);
        *(float4*)(dst0 + 4) = make_float4(d0[4], d0[5], d0[6], d0[7]);
        *(float4*)(dst1)     = make_float4(d1[0], d1[1], d1[2], d1[3]);
        *(float4*)(dst1 + 4) = make_float4(d1[4], d1[5], d1[6], d1[7]);
    }
}

extern "C" void kernel_launch(void* const* d_in, const int* in_sizes, int n_in,
                              void* d_out, int out_size, void* d_ws, size_t ws_size,
                              hipStream_t stream) {
    const float* x = (const float*)d_in[0];   // [2,16,64,64,64] fp32
    const float* w = (const float*)d_in[1];   // [1,32,16,3,3,3] fp32
    float* out = (float*)d_out;               // [2,32,64,64,64] fp32
    dim3 grid(NBLK), block(WPB * 32);
    hipLaunchKernelGGL(sparse_conv3d_wmma_kernel, grid, block, 0, stream, x, w, out);
}